// AdaptiveDiscriminatorAugmentation_38276748542555
// MI455X (gfx1250) — compile-verified
//
#include <hip/hip_runtime.h>
#include <hip/hip_bf16.h>

#define Bn 32
#define NCh 3
#define Hn 256
#define Wn 256
#define FWn 12
#define OSn 524                     // (256 + 2*3) * 2
#define MAXPAD 255
#define HPMAX (Hn + 2*MAXPAD)       // 766
#define HUPMAX (2*HPMAX)            // 1532

// workspace layout (floats), reused across batches (stream-serialized)
#define COEFF_OFF 0
#define COEFF_SZ  256
#define TMPV_OFF  (COEFF_OFF + COEFF_SZ)
#define TMPV_SZ   (NCh*HUPMAX*HPMAX)          // 3.52M fl
#define XUP_OFF   (TMPV_OFF + TMPV_SZ)
#define XUP_SZ    (NCh*HUPMAX*HUPMAX)         // 7.04M fl
#define XS_OFF    (XUP_OFF + XUP_SZ)
#define XS_SZ     (NCh*OSn*OSn)               // 824k fl
#define TMPD_OFF  (XS_OFF + XS_SZ)
#define TMPD_SZ   (NCh*Hn*OSn)                // 402k fl
// total ~ 11.8M floats ~ 47.2 MB

typedef float v2f __attribute__((ext_vector_type(2)));
typedef float v8f __attribute__((ext_vector_type(8)));

// ---------------- per-batch affine composition (G chain folded to pixel-space) ----
__global__ void ada_prep(const float* __restrict__ Ginv,
                         const int* __restrict__ pmx0, const int* __restrict__ pmy0,
                         const int* __restrict__ pmx1, const int* __restrict__ pmy1,
                         float* __restrict__ coeff) {
  int b = threadIdx.x;
  if (b >= Bn) return;
  const int mx0 = *pmx0, my0 = *pmy0, mx1 = *pmx1, my1 = *pmy1;
  const int hp = Hn + my0 + my1, wp = Wn + mx0 + mx1;
  const float wup = 2.f * (float)wp, hup = 2.f * (float)hp;
  float A[9];
#pragma unroll
  for (int i = 0; i < 9; ++i) A[i] = Ginv[b * 9 + i];
  // G = T((mx0-mx1)/2,(my0-my1)/2) @ G
  {
    float tx = 0.5f * (float)(mx0 - mx1), ty = 0.5f * (float)(my0 - my1);
    A[0] += tx * A[6]; A[1] += tx * A[7]; A[2] += tx * A[8];
    A[3] += ty * A[6]; A[4] += ty * A[7]; A[5] += ty * A[8];
  }
  // G = S(2,2) @ G @ S(.5,.5)
  A[2] *= 2.f; A[5] *= 2.f; A[6] *= 0.5f; A[7] *= 0.5f;
  // G = T(-.5,-.5) @ G @ T(.5,.5)   (right-mult first, then left-mult)
  A[2] += 0.5f * (A[0] + A[1]); A[5] += 0.5f * (A[3] + A[4]); A[8] += 0.5f * (A[6] + A[7]);
  A[0] -= 0.5f * A[6]; A[1] -= 0.5f * A[7]; A[2] -= 0.5f * A[8];
  A[3] -= 0.5f * A[6]; A[4] -= 0.5f * A[7]; A[5] -= 0.5f * A[8];
  // G = S(2/wup,2/hup) @ G @ S(wo/2,ho/2)
  {
    float sxl = 2.f / wup, syl = 2.f / hup, sr = (float)OSn * 0.5f;
    A[0] *= sxl * sr; A[1] *= sxl * sr; A[2] *= sxl;
    A[3] *= syl * sr; A[4] *= syl * sr; A[5] *= syl;
  }
  // fold grid->sample: p = (theta·[u,v,1] + 1)*dim/2 - 0.5
  coeff[b * 8 + 0] = A[0] * wup * 0.5f;
  coeff[b * 8 + 1] = A[1] * wup * 0.5f;
  coeff[b * 8 + 2] = (A[2] + 1.f) * wup * 0.5f - 0.5f;
  coeff[b * 8 + 3] = A[3] * hup * 0.5f;
  coeff[b * 8 + 4] = A[4] * hup * 0.5f;
  coeff[b * 8 + 5] = (A[5] + 1.f) * hup * 0.5f - 0.5f;
}

// ------------- vertical polyphase 2x upsample, reflect-pad fused --------------
__global__ void ada_up_v(const float* __restrict__ img, const float* __restrict__ Hz,
                         const int* pmx0, const int* pmy0, const int* pmx1, const int* pmy1,
                         float* __restrict__ tmpv, int b) {
  const int mx0 = *pmx0, my0 = *pmy0, mx1 = *pmx1, my1 = *pmy1;
  const int hp = Hn + my0 + my1, wp = Wn + mx0 + mx1, hup = 2 * hp;
  int idx = blockIdx.x * blockDim.x + threadIdx.x;
  if (idx >= NCh * hup * wp) return;
  int x = idx % wp; int t = idx / wp; int yy = t % hup; int c = t / hup;
  float ge[6], go[6];
#pragma unroll
  for (int k = 0; k < 6; ++k) { ge[k] = Hz[2 * k]; go[k] = Hz[2 * k + 1]; }
  int xs = x - mx0; xs = xs < 0 ? -xs : xs; xs = xs >= Wn ? 2 * Wn - 2 - xs : xs;
  const float* src = img + ((size_t)(b * NCh + c) * Hn) * Wn + xs;
  int base = yy + 5; int par = base & 1; int j0 = (base - par) >> 1;
  float acc = 0.f;
#pragma unroll
  for (int tt = 0; tt < 6; ++tt) {
    int j = j0 - tt;
    if (j >= 0 && j < hp) {
      int r = j - my0; r = r < 0 ? -r : r; r = r >= Hn ? 2 * Hn - 2 - r : r;
      float wgt = par ? go[tt] : ge[tt];
      acc += wgt * src[(size_t)r * Wn];
    }
  }
  tmpv[((size_t)c * hup + yy) * wp + x] = 2.f * acc;   // sqrt(gain)=2 per pass
}

// ------------- horizontal polyphase 2x upsample --------------
__global__ void ada_up_h(const float* __restrict__ tmpv, const float* __restrict__ Hz,
                         const int* pmx0, const int* pmy0, const int* pmx1, const int* pmy1,
                         float* __restrict__ xup) {
  const int mx0 = *pmx0, my0 = *pmy0, mx1 = *pmx1, my1 = *pmy1;
  const int hp = Hn + my0 + my1, wp = Wn + mx0 + mx1, hup = 2 * hp, wupi = 2 * wp;
  int idx = blockIdx.x * blockDim.x + threadIdx.x;
  if (idx >= NCh * hup * wupi) return;
  int xx = idx % wupi; int t = idx / wupi; int yy = t % hup; int c = t / hup;
  float ge[6], go[6];
#pragma unroll
  for (int k = 0; k < 6; ++k) { ge[k] = Hz[2 * k]; go[k] = Hz[2 * k + 1]; }
  const float* srcrow = tmpv + ((size_t)c * hup + yy) * wp;
  int base = xx + 5; int par = base & 1; int j0 = (base - par) >> 1;
  float acc = 0.f;
#pragma unroll
  for (int tt = 0; tt < 6; ++tt) {
    int j = j0 - tt;
    if (j >= 0 && j < wp) {
      float wgt = par ? go[tt] : ge[tt];
      acc += wgt * srcrow[j];
    }
  }
  xup[((size_t)c * hup + yy) * wupi + xx] = 2.f * acc;
}

// ------------- affine bilinear warp onto 524x524 grid --------------
__global__ void ada_sample(const float* __restrict__ xup, const float* __restrict__ coeff,
                           const int* pmx0, const int* pmy0, const int* pmx1, const int* pmy1,
                           float* __restrict__ xsb, int b) {
  const int mx0 = *pmx0, my0 = *pmy0, mx1 = *pmx1, my1 = *pmy1;
  const int hp = Hn + my0 + my1, wp = Wn + mx0 + mx1, hup = 2 * hp, wupi = 2 * wp;
  int idx = blockIdx.x * blockDim.x + threadIdx.x;
  if (idx >= NCh * OSn * OSn) return;
  int ox = idx % OSn; int t = idx / OSn; int oy = t % OSn; int c = t / OSn;
  float ax = coeff[b * 8 + 0], bx = coeff[b * 8 + 1], cx = coeff[b * 8 + 2];
  float ay = coeff[b * 8 + 3], by = coeff[b * 8 + 4], cy = coeff[b * 8 + 5];
  float u = ((float)ox + 0.5f) * (2.f / (float)OSn) - 1.f;
  float v = ((float)oy + 0.5f) * (2.f / (float)OSn) - 1.f;
  float px = ax * u + bx * v + cx;
  float py = ay * u + by * v + cy;
  float x0f = floorf(px), y0f = floorf(py);
  int ix0 = (int)x0f, iy0 = (int)y0f;
  float wx = px - x0f, wy = py - y0f;
  const float* plane = xup + (size_t)c * hup * wupi;
  float acc = 0.f;
  if (iy0 >= 0 && iy0 < hup) {
    if (ix0 >= 0 && ix0 < wupi)         acc += (1.f - wy) * (1.f - wx) * plane[(size_t)iy0 * wupi + ix0];
    if (ix0 + 1 >= 0 && ix0 + 1 < wupi) acc += (1.f - wy) * wx         * plane[(size_t)iy0 * wupi + ix0 + 1];
  }
  if (iy0 + 1 >= 0 && iy0 + 1 < hup) {
    if (ix0 >= 0 && ix0 < wupi)         acc += wy * (1.f - wx) * plane[(size_t)(iy0 + 1) * wupi + ix0];
    if (ix0 + 1 >= 0 && ix0 + 1 < wupi) acc += wy * wx         * plane[(size_t)(iy0 + 1) * wupi + ix0 + 1];
  }
  xsb[((size_t)c * OSn + oy) * OSn + ox] = acc;
}

// ------------- vertical 12-tap stride-2 downsample --------------
__global__ void ada_down_v(const float* __restrict__ xsb, const float* __restrict__ Hz,
                           float* __restrict__ tmpd) {
  int idx = blockIdx.x * blockDim.x + threadIdx.x;
  if (idx >= NCh * Hn * OSn) return;
  int x = idx % OSn; int t = idx / OSn; int y = t % Hn; int c = t / Hn;
  const float* plane = xsb + (size_t)c * OSn * OSn;
  float acc = 0.f;
#pragma unroll
  for (int k = 0; k < FWn; ++k) acc += Hz[k] * plane[(size_t)(2 * y + 1 + k) * OSn + x];
  tmpd[((size_t)c * Hn + y) * OSn + x] = acc;
}

// ------------- horizontal downsample: async-to-LDS row staging (CDNA5) ----------
__global__ void ada_down_h(const float* __restrict__ tmpd, const float* __restrict__ Hz,
                           float* __restrict__ out, int b) {
  __shared__ float srow[522];
  int c = blockIdx.x / Hn;
  int y = blockIdx.x % Hn;
  const float* srcrow = tmpd + ((size_t)c * Hn + y) * OSn;
  // stage srcrow[1..522] into LDS via async global->LDS copies (ASYNCcnt tracked)
  for (int i = threadIdx.x; i < 522; i += blockDim.x) {
    unsigned lo = (unsigned)(unsigned long long)(&srow[i]);
    unsigned long long ga = (unsigned long long)(srcrow + 1 + i);
    asm volatile("global_load_async_to_lds_b32 %0, %1, off" :: "v"(lo), "v"(ga) : "memory");
  }
  asm volatile("s_wait_asynccnt 0" ::: "memory");
  __syncthreads();
  int x = threadIdx.x;
  float acc = 0.f;
#pragma unroll
  for (int k = 0; k < FWn; ++k) acc += Hz[k] * srow[2 * x + k];   // stride-2: conflict-free on 64 banks
  out[(((size_t)b * NCh + c) * Hn + y) * Wn + x] = acc;
}

// ------------- color transform as f32 WMMA 16x16x4: D = C3x3·x + bias ----------
// A (16x4): rows 0..2 = color matrix rows (incl. bias col 3), rest 0
// B (4x16): rows 0..2 = 3 channels of 16 pixels, row 3 = ones (bias)
// Branchless operand gathers (address cndmask), 16 groups per wave to amortize A.
__global__ void ada_color(const float* __restrict__ Cmat, float* __restrict__ out) {
  int gtid = blockIdx.x * blockDim.x + threadIdx.x;
  int wave = gtid >> 5;
  int lane = threadIdx.x & 31;
  int b = wave >> 8;                 // 256 waves per batch
  int g0 = (wave & 255) << 8;        // 256 pixels per wave (16 groups of 16)
  int n = lane & 15;
  bool hi = lane >= 16;
  float* base = out + (size_t)b * NCh * Hn * Wn;
  const float* Cb = Cmat + b * 16;
  // ISA 7.12.2: A/B VGPR0 holds K0 (lanes 0-15) / K2 (lanes 16-31); VGPR1 K1/K3
  int cn = (n < 2) ? n : 2;          // clamp row index: keep loads in-bounds for all lanes
  v2f av;
  av.x = Cb[cn * 4 + (hi ? 2 : 0)];
  av.y = Cb[cn * 4 + (hi ? 3 : 1)];
  if (n >= 3) { av.x = 0.f; av.y = 0.f; }   // rows 3..15 of A are zero (cndmask, no branch)
#pragma unroll
  for (int it = 0; it < 16; ++it) {
    int p = g0 + it * 16 + n;
    // branchless B gather: one load each, address selected per lane half
    const float* a0 = base + (hi ? (131072 + p) : p);   // ch2 for hi lanes, ch0 for lo
    float b0 = *a0;
    float b1 = base[65536 + p];                         // ch1 (valid addr for all lanes)
    v2f bv;
    bv.x = b0;
    bv.y = hi ? 1.0f : b1;                              // ones row -> bias via K=3
    v8f acc = {0.f, 0.f, 0.f, 0.f, 0.f, 0.f, 0.f, 0.f};
    acc = __builtin_amdgcn_wmma_f32_16x16x4_f32(false, av, false, bv, (short)0, acc,
                                                false, false);
    // D VGPR v, lanes 0-15 -> row M=v: rows 0..2 = colored channels of pixel n
    if (!hi) {
      base[p]          = acc[0];
      base[65536 + p]  = acc[1];
      base[131072 + p] = acc[2];
    }
  }
}

extern "C" void kernel_launch(void* const* d_in, const int* in_sizes, int n_in,
                              void* d_out, int out_size, void* d_ws, size_t ws_size,
                              hipStream_t stream) {
  (void)in_sizes; (void)n_in; (void)out_size; (void)ws_size;
  const float* images = (const float*)d_in[0];
  const float* Ginv   = (const float*)d_in[1];
  const float* Cmat   = (const float*)d_in[2];
  const float* Hz     = (const float*)d_in[3];
  const int* mx0 = (const int*)d_in[4];
  const int* my0 = (const int*)d_in[5];
  const int* mx1 = (const int*)d_in[6];
  const int* my1 = (const int*)d_in[7];
  float* out = (float*)d_out;
  float* ws  = (float*)d_ws;
  float* coeff = ws + COEFF_OFF;
  float* tmpv  = ws + TMPV_OFF;
  float* xup   = ws + XUP_OFF;
  float* xsb   = ws + XS_OFF;
  float* tmpd  = ws + TMPD_OFF;

  ada_prep<<<1, 32, 0, stream>>>(Ginv, mx0, my0, mx1, my1, coeff);

  const int B1V = (NCh * HUPMAX * HPMAX + 255) / 256;   // worst-case grids; threads guard
  const int B1H = (NCh * HUPMAX * HUPMAX + 255) / 256;
  const int B2  = (NCh * OSn * OSn + 255) / 256;
  const int B3V = (NCh * Hn * OSn + 255) / 256;
  for (int b = 0; b < Bn; ++b) {    // per-batch pipeline: ws reused, L2-resident
    ada_up_v  <<<B1V, 256, 0, stream>>>(images, Hz, mx0, my0, mx1, my1, tmpv, b);
    ada_up_h  <<<B1H, 256, 0, stream>>>(tmpv, Hz, mx0, my0, mx1, my1, xup);
    ada_sample<<<B2,  256, 0, stream>>>(xup, coeff, mx0, my0, mx1, my1, xsb, b);
    ada_down_v<<<B3V, 256, 0, stream>>>(xsb, Hz, tmpd);
    ada_down_h<<<NCh * Hn, 256, 0, stream>>>(tmpd, Hz, out, b);
  }
  // 32 batches * 256 waves * 32 threads / 256 = 1024 blocks (exact; EXEC full at wmma)
  ada_color<<<1024, 256, 0, stream>>>(Cmat, out);
}